// DGCNN_52802327937626
// MI455X (gfx1250) — compile-verified
//
#include <hip/hip_runtime.h>
#include <hip/hip_bf16.h>

typedef float v2f __attribute__((ext_vector_type(2)));
typedef float v8f __attribute__((ext_vector_type(8)));

#define BATCH 16
#define NPTS  2048
#define KNN   20
#define CTOT  320
#define EPSV  1e-5f

// ---------------------------------------------------------------------------
// fp32 WMMA 16x16x4:  D(16x16,f32) = A(16x4,f32) * B(4x16,f32) + C
// A: lane<16 -> row=lane, v0=K0,v1=K1 ; lane>=16 -> row=lane-16, v0=K2,v1=K3
// B: lane<16 -> col=lane, v0=K0,v1=K1 ; lane>=16 -> col=lane-16, v0=K2,v1=K3
// C/D: element (M = vgpr + 8*(lane>=16), N = lane&15)
// ---------------------------------------------------------------------------
__device__ __forceinline__ v8f wmma4(v2f a, v2f b, v8f c) {
  return __builtin_amdgcn_wmma_f32_16x16x4_f32(false, a, false, b, (short)0, c,
                                               false, false);
}

// Exact (order-independent) float atomic max/min via int/uint monotone trick.
// Requires init to -inf / +inf respectively.
__device__ __forceinline__ void atomicMaxF(float* a, float v) {
  if (v >= 0.0f) atomicMax((int*)a, __float_as_int(v));
  else           atomicMin((unsigned int*)a, __float_as_uint(v));
}
__device__ __forceinline__ void atomicMinF(float* a, float v) {
  if (v >= 0.0f) atomicMin((int*)a, __float_as_int(v));
  else           atomicMax((unsigned int*)a, __float_as_uint(v));
}

// ---------------------------------------------------------------------------
// sq[b,n] = sum_c x[b,c,n]^2
// ---------------------------------------------------------------------------
__global__ void sq_kernel(const float* __restrict__ x, int cbs, int coff, int C,
                          float* __restrict__ sq) {
  int t = blockIdx.x * blockDim.x + threadIdx.x;
  if (t >= BATCH * NPTS) return;
  int bb = t / NPTS, n = t % NPTS;
  const float* xb = x + ((size_t)bb * cbs + coff) * NPTS + n;
  float s = 0.f;
  for (int c = 0; c < C; ++c) { float v = xb[(size_t)c * NPTS]; s += v * v; }
  sq[t] = s;
}

// ---------------------------------------------------------------------------
// d2[n,m] = sq[n] + sq[m] - 2 * sum_c x[c,n]*x[c,m]   (one batch)
// One wave = one 16x16 tile; K-loop over channels in steps of 4 (WMMA fp32).
// Templated on C so the C%4==0 path has no guards / no divergent branches.
// ---------------------------------------------------------------------------
template <int C>
__global__ __launch_bounds__(128) void d2_gemm_kernel(
    const float* __restrict__ x, int cbs, int coff,
    const float* __restrict__ sq, float* __restrict__ d2, int bb) {
  const int lane = threadIdx.x & 31, wid = threadIdx.x >> 5;
  const int w = blockIdx.x * 4 + wid;          // 0..16383
  const int tn = w >> 7, tm = w & 127;
  const int half = lane >> 4, l15 = lane & 15;
  const int n = tn * 16 + l15;
  const int m = tm * 16 + l15;
  const float* xb = x + ((size_t)bb * cbs + coff) * NPTS;
  v8f acc = {};
  if constexpr ((C & 3) == 0) {
#pragma unroll 4
    for (int ci = 0; ci < C / 4; ++ci) {
      int c0 = ci * 4 + half * 2;
      v2f a, bv;
      a.x  = xb[(size_t)c0       * NPTS + n];
      a.y  = xb[(size_t)(c0 + 1) * NPTS + n];
      bv.x = xb[(size_t)c0       * NPTS + m];
      bv.y = xb[(size_t)(c0 + 1) * NPTS + m];
      acc = wmma4(a, bv, acc);
    }
  } else {
#pragma unroll
    for (int ci = 0; ci < (C + 3) / 4; ++ci) {
      int c0 = ci * 4 + half * 2;
      v2f a, bv;
      a.x  = (c0     < C) ? xb[(size_t)c0       * NPTS + n] : 0.f;
      a.y  = (c0 + 1 < C) ? xb[(size_t)(c0 + 1) * NPTS + n] : 0.f;
      bv.x = (c0     < C) ? xb[(size_t)c0       * NPTS + m] : 0.f;
      bv.y = (c0 + 1 < C) ? xb[(size_t)(c0 + 1) * NPTS + m] : 0.f;
      acc = wmma4(a, bv, acc);
    }
  }
  float sqm = sq[bb * NPTS + tm * 16 + l15];
#pragma unroll
  for (int r = 0; r < 8; ++r) {
    int row = tn * 16 + r + half * 8;
    d2[(size_t)row * NPTS + tm * 16 + l15] =
        sq[bb * NPTS + row] + sqm - 2.f * acc[r];
  }
}

// ---------------------------------------------------------------------------
// Stable top-K (K=20 smallest d2, ties -> lower index). One wave per query.
// 20 lexicographic-min selection passes over register-cached row.
// ---------------------------------------------------------------------------
__global__ void topk_kernel(const float* __restrict__ d2, int* __restrict__ idx,
                            int bb) {
  const int lane = threadIdx.x & 31, wid = threadIdx.x >> 5;
  const int n = blockIdx.x * 4 + wid;          // 0..2047
  const float* row = d2 + (size_t)n * NPTS;
  float v[64];
#pragma unroll
  for (int j = 0; j < 64; ++j) v[j] = row[lane + 32 * j];
  float tv = -__builtin_inff();
  int   ti = -1;
  int* out = idx + ((size_t)bb * NPTS + n) * KNN;
  for (int p = 0; p < KNN; ++p) {
    float bv = __builtin_inff();
    int   bi = 0x7fffffff;
#pragma unroll
    for (int j = 0; j < 64; ++j) {
      int m = lane + 32 * j;
      float vv = v[j];
      bool elig = (vv > tv) || (vv == tv && m > ti);   // strictly after thresh
      if (elig && (vv < bv || (vv == bv && m < bi))) { bv = vv; bi = m; }
    }
    for (int off = 16; off; off >>= 1) {
      float ov = __shfl_xor(bv, off, 32);
      int   oi = __shfl_xor(bi, off, 32);
      if (ov < bv || (ov == bv && oi < bi)) { bv = ov; bi = oi; }
    }
    tv = bv; ti = bi;
    if (lane == 0) out[p] = bi;
  }
}

// ---------------------------------------------------------------------------
__global__ void init_stats_kernel(float* ymax, float* ymin, int nelem) {
  int t = blockIdx.x * blockDim.x + threadIdx.x;
  if (t < nelem) { ymax[t] = -__builtin_inff(); ymin[t] = __builtin_inff(); }
}

// ---------------------------------------------------------------------------
// Edge-conv GEMM: y[o,b,n,k] = sum_c W[o,c]*h[c] + bias[o]
//   h[c] = c<C ? x[c,n] : x[c-C,m]-x[c-C,n],  m = idx[b,n,k]
// Reduces max_k/min_k into ymax/ymin (atomic, exact) and per-channel
// sum/sumsq into fixed per-wave partial slots (deterministic).
// Templated on C: C%4==0 -> split ctr/nbr loops, zero per-lane branching.
// ---------------------------------------------------------------------------
template <int C>
__global__ __launch_bounds__(128) void ec_gemm_kernel(
    const float* __restrict__ x, int cbs, int coff,
    const float* __restrict__ W, const float* __restrict__ bias,
    const int* __restrict__ idx, int O,
    float* __restrict__ ymax, float* __restrict__ ymin,
    float* __restrict__ psum, float* __restrict__ psq) {
  const int lane = threadIdx.x & 31, wid = threadIdx.x >> 5;
  const int half = lane >> 4, l15 = lane & 15;
  const int ot = blockIdx.y;
  const int twoC = 2 * C;
  const int NT = (BATCH * NPTS * KNN) / 16;        // 40960 sample tiles
  const int wslot = blockIdx.x * 4 + wid;          // 0..2559
  const int orow = ot * 16 + l15;
  float sum[8] = {}, sumsq[8] = {};
  for (int t = wslot; t < NT; t += gridDim.x * 4) {
    const int s   = t * 16 + l15;                  // flat (b,n,k)
    const int bb  = s / (NPTS * KNN);
    const int rem = s - bb * (NPTS * KNN);
    const int n   = rem / KNN;
    const int m   = idx[s];
    const float* xb = x + ((size_t)bb * cbs + coff) * NPTS;
    v8f acc = {};
    if constexpr ((C & 3) == 0) {
      // center half: h[c] = x[c,n]
#pragma unroll 4
      for (int ci = 0; ci < C / 4; ++ci) {
        int c0 = ci * 4 + half * 2;
        v2f a, bv;
        a.x  = W[orow * twoC + c0];
        a.y  = W[orow * twoC + c0 + 1];
        bv.x = xb[(size_t)c0       * NPTS + n];
        bv.y = xb[(size_t)(c0 + 1) * NPTS + n];
        acc = wmma4(a, bv, acc);
      }
      // neighbor half: h[C+c] = x[c,m] - x[c,n]
#pragma unroll 4
      for (int ci = 0; ci < C / 4; ++ci) {
        int c0 = ci * 4 + half * 2;
        v2f a, bv;
        a.x  = W[orow * twoC + C + c0];
        a.y  = W[orow * twoC + C + c0 + 1];
        bv.x = xb[(size_t)c0       * NPTS + m] - xb[(size_t)c0       * NPTS + n];
        bv.y = xb[(size_t)(c0 + 1) * NPTS + m] - xb[(size_t)(c0 + 1) * NPTS + n];
        acc = wmma4(a, bv, acc);
      }
    } else {
#pragma unroll
      for (int ci = 0; ci < (twoC + 3) / 4; ++ci) {
        int c0 = ci * 4 + half * 2;
        v2f a, bv;
        a.x = (c0     < twoC) ? W[orow * twoC + c0]     : 0.f;
        a.y = (c0 + 1 < twoC) ? W[orow * twoC + c0 + 1] : 0.f;
        float h0 = 0.f, h1 = 0.f;
        if (c0 < C)                          h0 = xb[(size_t)c0 * NPTS + n];
        else if (c0 < twoC)                  h0 = xb[(size_t)(c0 - C) * NPTS + m]
                                                - xb[(size_t)(c0 - C) * NPTS + n];
        if (c0 + 1 < C)                      h1 = xb[(size_t)(c0 + 1) * NPTS + n];
        else if (c0 + 1 < twoC)              h1 = xb[(size_t)(c0 + 1 - C) * NPTS + m]
                                                - xb[(size_t)(c0 + 1 - C) * NPTS + n];
        bv.x = h0; bv.y = h1;
        acc = wmma4(a, bv, acc);
      }
    }
#pragma unroll
    for (int r = 0; r < 8; ++r) {
      int o = ot * 16 + r + half * 8;
      float y = acc[r] + bias[o];
      sum[r] += y; sumsq[r] += y * y;
      size_t e = ((size_t)bb * O + o) * NPTS + n;
      atomicMaxF(&ymax[e], y);
      atomicMinF(&ymin[e], y);
    }
  }
#pragma unroll
  for (int r = 0; r < 8; ++r) {
    float s1 = sum[r], s2 = sumsq[r];
    for (int msk = 1; msk <= 8; msk <<= 1) {       // stays within 16-lane half
      s1 += __shfl_xor(s1, msk, 32);
      s2 += __shfl_xor(s2, msk, 32);
    }
    if (l15 == 0) {
      int o = ot * 16 + r + half * 8;
      psum[(size_t)o * 2560 + wslot] = s1;
      psq [(size_t)o * 2560 + wslot] = s2;
    }
  }
}

// Fixed-order reduction of per-wave partials -> chs/chsq (deterministic).
__global__ void reduce_stats_kernel(const float* __restrict__ psum,
                                    const float* __restrict__ psq, int nslots,
                                    int O, float* chs, float* chsq) {
  int o = blockIdx.x * blockDim.x + threadIdx.x;
  if (o >= O) return;
  float s = 0.f, q = 0.f;
  for (int i = 0; i < nslots; ++i) {
    s += psum[(size_t)o * nslots + i];
    q += psq [(size_t)o * nslots + i];
  }
  chs[o] = s; chsq[o] = q;
}

// BN (folded) + ReLU applied to max_k / min_k -> feature slice of xcat.
__global__ void ec_final_kernel(const float* __restrict__ ymax,
                                const float* __restrict__ ymin,
                                const float* __restrict__ chs,
                                const float* __restrict__ chsq,
                                const float* __restrict__ g,
                                const float* __restrict__ be, int O, int outcoff,
                                float* __restrict__ xcat) {
  int t = blockIdx.x * blockDim.x + threadIdx.x;
  int total = BATCH * O * NPTS;
  if (t >= total) return;
  int n  = t % NPTS;
  int o  = (t / NPTS) % O;
  int bb = t / (NPTS * O);
  const float cnt = (float)(BATCH * NPTS * KNN);
  float mean  = chs[o] / cnt;
  float var   = chsq[o] / cnt - mean * mean;
  float scale = g[o] * rsqrtf(var + EPSV);
  float shift = be[o] - mean * scale;
  float base = (scale >= 0.f) ? ymax[t] : ymin[t];
  float v = scale * base + shift;
  xcat[((size_t)bb * CTOT + outcoff + o) * NPTS + n] = fmaxf(v, 0.f);
}

// ---------------------------------------------------------------------------
// 1x1 conv 320->1024 over all points, with running max/argmax (first occ.)
// and min/argmin per (b,o), plus per-channel sum/sumsq partials.
// One wave per (o-tile, batch); W tile staged through LDS (stride 321 pad).
// ---------------------------------------------------------------------------
__global__ __launch_bounds__(128) void mlp_kernel(
    const float* __restrict__ xcat, const float* __restrict__ W,
    const float* __restrict__ bias,
    float* __restrict__ gmv, int* __restrict__ gmn,
    float* __restrict__ gnv, int* __restrict__ gnn,
    float* __restrict__ psum, float* __restrict__ psq) {
  __shared__ float Wl[16 * 321];
  const int ot = blockIdx.x;                // 0..63
  const int o0 = ot * 16;
  for (int i = threadIdx.x; i < 16 * 320; i += 128) {
    int r = i / 320, c = i - r * 320;
    Wl[r * 321 + c] = W[(o0 + r) * 320 + c];
  }
  __syncthreads();
  const int lane = threadIdx.x & 31, wid = threadIdx.x >> 5;
  const int half = lane >> 4, l15 = lane & 15;
  const int bb = blockIdx.y * 4 + wid;      // 0..15
  float mv[8], nv_[8], sum[8] = {}, sumsq[8] = {}, bsv[8];
  int mn[8], nn_[8];
#pragma unroll
  for (int r = 0; r < 8; ++r) {
    mv[r] = -__builtin_inff(); nv_[r] = __builtin_inff();
    mn[r] = 0; nn_[r] = 0;
    bsv[r] = bias[o0 + r + half * 8];
  }
  for (int st = 0; st < NPTS / 16; ++st) {
    const int n = st * 16 + l15;
    v8f acc = {};
#pragma unroll 4
    for (int ci = 0; ci < 80; ++ci) {       // 320 / 4
      int c0 = ci * 4 + half * 2;
      v2f a, bv;
      a.x  = Wl[l15 * 321 + c0];
      a.y  = Wl[l15 * 321 + c0 + 1];
      bv.x = xcat[((size_t)bb * CTOT + c0)     * NPTS + n];
      bv.y = xcat[((size_t)bb * CTOT + c0 + 1) * NPTS + n];
      acc = wmma4(a, bv, acc);
    }
#pragma unroll
    for (int r = 0; r < 8; ++r) {
      float y = acc[r] + bsv[r];
      sum[r] += y; sumsq[r] += y * y;
      if (y > mv[r])  { mv[r]  = y; mn[r]  = n; }   // strict: keeps first
      if (y < nv_[r]) { nv_[r] = y; nn_[r] = n; }
    }
  }
#pragma unroll
  for (int r = 0; r < 8; ++r) {
    float a_mv = mv[r];  int a_mn = mn[r];
    float a_nv = nv_[r]; int a_nn = nn_[r];
    float s1 = sum[r], s2 = sumsq[r];
    for (int msk = 1; msk <= 8; msk <<= 1) {
      float ov = __shfl_xor(a_mv, msk, 32); int oi = __shfl_xor(a_mn, msk, 32);
      if (ov > a_mv || (ov == a_mv && oi < a_mn)) { a_mv = ov; a_mn = oi; }
      float ov2 = __shfl_xor(a_nv, msk, 32); int oi2 = __shfl_xor(a_nn, msk, 32);
      if (ov2 < a_nv || (ov2 == a_nv && oi2 < a_nn)) { a_nv = ov2; a_nn = oi2; }
      s1 += __shfl_xor(s1, msk, 32);
      s2 += __shfl_xor(s2, msk, 32);
    }
    if (l15 == 0) {
      int o = o0 + r + half * 8;
      gmv[bb * 1024 + o] = a_mv; gmn[bb * 1024 + o] = a_mn;
      gnv[bb * 1024 + o] = a_nv; gnn[bb * 1024 + o] = a_nn;
      psum[(size_t)o * 16 + bb] = s1;
      psq [(size_t)o * 16 + bb] = s2;
    }
  }
}

// Final BN+ReLU on the global max; emit (value, argmax) per (b,o).
__global__ void out_kernel(const float* __restrict__ gmv,
                           const int* __restrict__ gmn,
                           const float* __restrict__ gnv,
                           const int* __restrict__ gnn,
                           const float* __restrict__ chs,
                           const float* __restrict__ chsq,
                           const float* __restrict__ g,
                           const float* __restrict__ be,
                           float* __restrict__ out) {
  int t = blockIdx.x * blockDim.x + threadIdx.x;
  if (t >= BATCH * 1024) return;
  int o = t % 1024;
  const float cnt = (float)(BATCH * NPTS);
  float mean  = chs[o] / cnt;
  float var   = chsq[o] / cnt - mean * mean;
  float scale = g[o] * rsqrtf(var + EPSV);
  float shift = be[o] - mean * scale;
  float v; int bi;
  if      (scale > 0.f) { v = scale * gmv[t] + shift; bi = gmn[t]; }
  else if (scale < 0.f) { v = scale * gnv[t] + shift; bi = gnn[t]; }
  else                  { v = shift; bi = 0; }
  out[t] = fmaxf(v, 0.f);
  out[BATCH * 1024 + t] = (float)((v > 0.f) ? bi : 0);   // relu<=0 => all-zero row, argmax=0
}

// ---------------------------------------------------------------------------
extern "C" void kernel_launch(void* const* d_in, const int* in_sizes, int n_in,
                              void* d_out, int out_size, void* d_ws,
                              size_t ws_size, hipStream_t stream) {
  (void)in_sizes; (void)n_in; (void)out_size; (void)ws_size;
  // jax tree_flatten order (sorted keys): params{ecs,local,tnet} then points.
  const float* ecW[4]  = {(const float*)d_in[0],  (const float*)d_in[4],
                          (const float*)d_in[8],  (const float*)d_in[12]};
  const float* ecB[4]  = {(const float*)d_in[1],  (const float*)d_in[5],
                          (const float*)d_in[9],  (const float*)d_in[13]};
  const float* ecBe[4] = {(const float*)d_in[2],  (const float*)d_in[6],
                          (const float*)d_in[10], (const float*)d_in[14]};
  const float* ecG[4]  = {(const float*)d_in[3],  (const float*)d_in[7],
                          (const float*)d_in[11], (const float*)d_in[15]};
  const float* locW  = (const float*)d_in[16];
  const float* locB  = (const float*)d_in[17];
  const float* locBe = (const float*)d_in[18];
  const float* locG  = (const float*)d_in[19];
  // d_in[20..41] = tnet params: dead code (lin_W == 0 => trans == I).
  const float* points = (const float*)d_in[42];

  float* ws   = (float*)d_ws;                 // ~77.4 MiB total
  float* xcat = ws;                           // 16*320*2048      = 10485760
  float* d2   = ws + 10485760;                // 2048*2048        =  4194304
  float* ymax = d2;                           // aliased: knn phase done first
  float* ymin = ws + 14680064;                //                  =  4194304
  float* sqb  = ws + 18874368;                // 16*2048          =    32768
  int*   idxb = (int*)(ws + 18907136);        // 16*2048*20       =   655360
  float* psum = ws + 19562496;                // 128*2560         =   327680
  float* psq  = ws + 19890176;                //                  =   327680
  float* chs  = ws + 20217856;                // 1024
  float* chsq = ws + 20218880;                // 1024
  float* gmv  = ws + 20219904;                // 16*1024
  int*   gmn  = (int*)(ws + 20236288);
  float* gnv  = ws + 20252672;
  int*   gnn  = (int*)(ws + 20269056);

  const int Cin[4]     = {3, 64, 64, 64};
  const int Oout[4]    = {64, 64, 64, 128};
  const int coffIn[4]  = {0, 0, 64, 128};
  const int coffOut[4] = {0, 64, 128, 192};

  for (int s = 0; s < 4; ++s) {
    const float* xin = (s == 0) ? points : xcat;
    const int cbs  = (s == 0) ? 3 : CTOT;
    const int coff = coffIn[s];
    const int C = Cin[s], O = Oout[s];

    sq_kernel<<<(BATCH * NPTS + 127) / 128, 128, 0, stream>>>(xin, cbs, coff, C,
                                                              sqb);
    for (int bb = 0; bb < BATCH; ++bb) {      // reuse one NxN d2 tile per batch
      if (s == 0)
        d2_gemm_kernel<3><<<4096, 128, 0, stream>>>(xin, cbs, coff, sqb, d2, bb);
      else
        d2_gemm_kernel<64><<<4096, 128, 0, stream>>>(xin, cbs, coff, sqb, d2, bb);
      topk_kernel<<<512, 128, 0, stream>>>(d2, idxb, bb);
    }
    const int nelem = O * BATCH * NPTS;
    init_stats_kernel<<<(nelem + 255) / 256, 256, 0, stream>>>(ymax, ymin, nelem);
    dim3 g1(640, O / 16);
    if (s == 0)
      ec_gemm_kernel<3><<<g1, 128, 0, stream>>>(xin, cbs, coff, ecW[s], ecB[s],
                                                idxb, O, ymax, ymin, psum, psq);
    else
      ec_gemm_kernel<64><<<g1, 128, 0, stream>>>(xin, cbs, coff, ecW[s], ecB[s],
                                                 idxb, O, ymax, ymin, psum, psq);
    reduce_stats_kernel<<<(O + 127) / 128, 128, 0, stream>>>(psum, psq, 2560, O,
                                                             chs, chsq);
    ec_final_kernel<<<(nelem + 255) / 256, 256, 0, stream>>>(
        ymax, ymin, chs, chsq, ecG[s], ecBe[s], O, coffOut[s], xcat);
  }

  dim3 gm(64, 4);
  mlp_kernel<<<gm, 128, 0, stream>>>(xcat, locW, locB, gmv, gmn, gnv, gnn, psum,
                                     psq);
  reduce_stats_kernel<<<(1024 + 127) / 128, 128, 0, stream>>>(psum, psq, 16,
                                                              1024, chs, chsq);
  out_kernel<<<(BATCH * 1024 + 255) / 256, 256, 0, stream>>>(
      gmv, gmn, gnv, gnn, chs, chsq, locG, locBe, (float*)d_out);
}